// FarthestPointSubSamplePC_72241349919056
// MI455X (gfx1250) — compile-verified
//
#include <hip/hip_runtime.h>
#include <stdint.h>

#define B_ 16
#define N_ 65536
#define C_ 6
#define K_ 1024
#define T_ 1024                       // threads per block (32 wave32 waves on one WGP)
#define PPT_ (N_ / T_)                // 64 points per thread
#define TILE_BYTES_ (T_ * C_ * 4)     // 24576 B of global points per tile step

__global__ __launch_bounds__(T_) void fps_kernel(
    const float* __restrict__ points, const int* __restrict__ start_idx,
    float* __restrict__ out_sampled, int* __restrict__ out_counts) {
  const int b    = blockIdx.x;
  const int tid  = threadIdx.x;
  const int lane = tid & 31;
  const int wid  = tid >> 5;

  const float* ptsb = points + (size_t)b * N_ * C_;

  // Double-buffered staging tiles: 2 x 16KB. Each thread owns 16B per tile.
  __shared__ __align__(16) float buf[2][T_ * 4];
  __shared__ float redv[T_ / 32];
  __shared__ int   redi[T_ / 32];
  __shared__ int   gsel_s;

  const unsigned lds_off0 = (unsigned)(uintptr_t)&buf[0][tid * 4];
  const unsigned lds_off1 = (unsigned)(uintptr_t)&buf[1][tid * 4];

  // Thread t owns global point indices { j*T_ + tid : j in [0,64) } (coalesced).
  const unsigned long long gbase =
      (unsigned long long)(uintptr_t)(ptsb + (size_t)tid * C_);

  float mind[PPT_];
#pragma unroll
  for (int j = 0; j < PPT_; ++j) mind[j] = __builtin_inff();

  int gsel = __builtin_amdgcn_readfirstlane(start_idx[b]);

  for (int k = 0; k < K_; ++k) {
    // Emit all C=6 channels of the current selection.
    if (tid < C_)
      out_sampled[((size_t)b * K_ + k) * C_ + tid] = ptsb[(size_t)gsel * C_ + tid];
    if (k == K_ - 1) break;

    const float sx = ptsb[(size_t)gsel * C_ + 0];
    const float sy = ptsb[(size_t)gsel * C_ + 1];
    const float sz = ptsb[(size_t)gsel * C_ + 2];

    float bestv  = -__builtin_inff();
    int   best_t = 0;   // tile counter only: 0..63 are inline constants -> no
                        // materialized index-constant VGPRs (seen in round-1 asm)

    // Preload tile 0 via CDNA5 async copy engine (ASYNCcnt-tracked).
    asm volatile("global_load_async_to_lds_b128 %0, %1, off"
                 :: "v"(lds_off0), "v"(gbase) : "memory");

#pragma unroll
    for (int t = 0; t < PPT_; ++t) {
      if (t + 1 < PPT_) {
        // Stage next tile into the other buffer while we consume this one.
        asm volatile("global_load_async_to_lds_b128 %0, %1, off offset:%2"
                     :: "v"(((t + 1) & 1) ? lds_off1 : lds_off0),
                        "v"(gbase),
                        "n"((t + 1) * TILE_BYTES_)
                     : "memory");
        // Async loads complete in order: cnt<=1 => tile t's LDS writes landed.
        asm volatile("s_wait_asynccnt 0x1" ::: "memory");
      } else {
        asm volatile("s_wait_asynccnt 0x0" ::: "memory");
      }
      const float4 p = *(const float4*)(&buf[t & 1][tid * 4]);  // ds_load_b128
      const float dx = p.x - sx;
      const float dy = p.y - sy;
      const float dz = p.z - sz;
      const float d  = dx * dx + dy * dy + dz * dz;
      const float m  = fminf(mind[t], d);
      mind[t] = m;
      // Strict '>' + increasing t order == jnp.argmax first-occurrence.
      if (m > bestv) { bestv = m; best_t = t; }
    }
    // Reconstruct full point index once (2 VALU instead of 64 live constants).
    int besti = best_t * T_ + tid;

    // wave32 argmax reduce (tie -> smaller index).
#pragma unroll
    for (int m = 16; m >= 1; m >>= 1) {
      const float ov = __shfl_xor(bestv, m, 32);
      const int   oi = __shfl_xor(besti, m, 32);
      if (ov > bestv || (ov == bestv && oi < besti)) { bestv = ov; besti = oi; }
    }
    if (lane == 0) { redv[wid] = bestv; redi[wid] = besti; }
    __syncthreads();
    if (wid == 0) {
      float v = redv[lane];
      int   i = redi[lane];
#pragma unroll
      for (int m = 16; m >= 1; m >>= 1) {
        const float ov = __shfl_xor(v, m, 32);
        const int   oi = __shfl_xor(i, m, 32);
        if (ov > v || (ov == v && oi < i)) { v = ov; i = oi; }
      }
      if (lane == 0) gsel_s = i;
    }
    __syncthreads();
    gsel = __builtin_amdgcn_readfirstlane(gsel_s);
  }

  if (tid == 0) out_counts[b] = K_;   // counts = full(K), int32 bit patterns
}

extern "C" void kernel_launch(void* const* d_in, const int* in_sizes, int n_in,
                              void* d_out, int out_size, void* d_ws, size_t ws_size,
                              hipStream_t stream) {
  (void)in_sizes; (void)n_in; (void)out_size; (void)d_ws; (void)ws_size;
  const float* points    = (const float*)d_in[0];
  const int*   start_idx = (const int*)d_in[1];
  float* out_sampled = (float*)d_out;
  int*   out_counts  = (int*)(out_sampled + (size_t)B_ * K_ * C_);
  fps_kernel<<<dim3(B_), dim3(T_), 0, stream>>>(points, start_idx,
                                                out_sampled, out_counts);
}